// Memory_25013889532427
// MI455X (gfx1250) — compile-verified
//
#include <hip/hip_runtime.h>
#include <hip/hip_bf16.h>
#include <math.h>

#define BS     256
#define HID    1024
#define ACTN   16
#define MEMD   512
#define NMEM   1024

// d_out layout (floats): read_v | other_v | M_new | alpha
#define READ_OFF   0
#define OTHER_OFF  (BS * MEMD)                        // 131072
#define M_OFF      (2 * BS * MEMD)                    // 262144
#define ALPHA_OFF  (M_OFF + (size_t)BS * NMEM * MEMD) // 262144 + 134217728

// ws layout (floats)
#define WS_ERASE 0
#define WS_ADD   (BS * MEMD)            // 131072
#define WS_GHID  (2 * BS * MEMD)        // 262144
#define WS_KERN  (3 * BS * MEMD)        // 393216
#define WS_RPART (3 * BS * MEMD + BS * 9) // 395520 (+ 256*4*512 = 524288)

typedef __attribute__((ext_vector_type(2))) float v2f;
typedef __attribute__((ext_vector_type(8))) float v8f;

// ---------------------------------------------------------------------------
// Kernel-net: per-batch 3x3 kernel = softmax( maybe-flipped( leaky(a@Wk1+bk1)@Wk2+bk2 ) )
// One block per batch row. Deterministic LDS tree reductions.
// ---------------------------------------------------------------------------
__global__ __launch_bounds__(256) void kernelnet_kernel(
    const float* __restrict__ a, const float* __restrict__ Wk1,
    const float* __restrict__ bk1, const float* __restrict__ Wk2,
    const float* __restrict__ bk2, float* __restrict__ kern9)
{
  const int b = blockIdx.x, tid = threadIdx.x;
  __shared__ float s_a[ACTN];
  __shared__ int   s_m;
  __shared__ float red[256];
  __shared__ float s_k[9];

  if (tid < ACTN) s_a[tid] = a[b * ACTN + tid];
  __syncthreads();
  if (tid == 0) {
    float best = s_a[0]; int bi = 0;
    for (int i = 1; i < ACTN; ++i) if (s_a[i] > best) { best = s_a[i]; bi = i; }
    s_m = (bi == 0);
  }
  __syncthreads();
  const int m = s_m;

  float au[ACTN];
  #pragma unroll
  for (int i = 0; i < ACTN; ++i) au[i] = s_a[i];
  if (m) { au[0] = 0.f; au[1] = 1.f; }     // new_a path (only needed variant)

  float p[9];
  #pragma unroll
  for (int o = 0; o < 9; ++o) p[o] = 0.f;
  for (int jj = 0; jj < 2; ++jj) {
    const int j = tid + jj * 256;          // 512 hidden units / 256 threads
    float kj = bk1[j];
    #pragma unroll
    for (int i = 0; i < ACTN; ++i) kj += au[i] * Wk1[i * MEMD + j];
    kj = kj > 0.f ? kj : 0.2f * kj;        // leaky_relu 0.2
    #pragma unroll
    for (int o = 0; o < 9; ++o) p[o] += kj * Wk2[j * 9 + o];
  }
  for (int o = 0; o < 9; ++o) {
    red[tid] = p[o]; __syncthreads();
    for (int s = 128; s > 0; s >>= 1) { if (tid < s) red[tid] += red[tid + s]; __syncthreads(); }
    if (tid == 0) s_k[o] = red[0] + bk2[o];
    __syncthreads();
  }
  if (tid == 0) {
    float kk[9];
    for (int o = 0; o < 9; ++o) kk[o] = m ? s_k[8 - o] : s_k[o];  // spatial flip if masked
    float mx = kk[0];
    for (int o = 1; o < 9; ++o) mx = fmaxf(mx, kk[o]);
    float e[9], sum = 0.f;
    for (int o = 0; o < 9; ++o) { e[o] = expf(kk[o] - mx); sum += e[o]; }
    const float inv = 1.f / sum;
    for (int o = 0; o < 9; ++o) kern9[b * 9 + o] = e[o] * inv;
  }
}

// ---------------------------------------------------------------------------
// FP32 WMMA GEMM: C(256 x N) = h(256 x 1024) @ W(1024 x N) + bias, fused epilogue.
//   mode 0: leaky_relu(x, 0.2) -> o_erase (== ghid, N = 512)
//   mode 1: col<512  : sigmoid -> o_erase
//           col<1024 : raw     -> o_add   (col-512)
//           else     : raw     -> o_other (col-1024)
// One wave = one 16x16 C tile; K marched in steps of 4 via v_wmma_f32_16x16x4_f32.
// A 16x4 layout: lane%16 = M row, VGPR0/1 = K {kb, kb+1}, kb = (lane/16)*2.
// B 4x16 layout: lane%16 = N col, VGPR0/1 = K {kb, kb+1}.
// C/D: VGPR r -> row r (lanes 0-15) / row r+8 (lanes 16-31), col = lane%16.
// ---------------------------------------------------------------------------
__global__ __launch_bounds__(128) void gemm_wmma_f32(
    const float* __restrict__ A, const float* __restrict__ W,
    const float* __restrict__ bias, int N, int mode,
    float* __restrict__ o_erase, float* __restrict__ o_add,
    float* __restrict__ o_other)
{
  const int K = HID;
  const int wave = threadIdx.x >> 5;
  const int lane = threadIdx.x & 31;
  const int t = blockIdx.x * 4 + wave;
  const int ntiles = (BS / 16) * (N / 16);
  if (t >= ntiles) return;                       // whole-wave uniform: EXEC stays full
  const int mt = t % (BS / 16);
  const int nt = t / (BS / 16);

  const int rc = lane & 15;                      // A row / B col within tile
  const int kb = (lane >> 4) << 1;               // K sub-offset for this half-wave
  const float* __restrict__ Arow = A + (size_t)(mt * 16 + rc) * K;
  const float* __restrict__ Wcol = W + nt * 16 + rc;

  v8f acc = {};
  #pragma unroll 4
  for (int k0 = 0; k0 < K; k0 += 4) {
    v2f av; av.x = Arow[k0 + kb];                 av.y = Arow[k0 + kb + 1];
    v2f bvv; bvv.x = Wcol[(size_t)(k0 + kb) * N]; bvv.y = Wcol[(size_t)(k0 + kb + 1) * N];
    acc = __builtin_amdgcn_wmma_f32_16x16x4_f32(
        /*neg_a=*/false, av, /*neg_b=*/false, bvv,
        /*c_mod=*/(short)0, acc, /*reuse_a=*/false, /*reuse_b=*/false);
  }

  const int gcol  = nt * 16 + rc;
  const int rbase = mt * 16 + ((lane >> 4) << 3);
  const float bb = bias[gcol];
  #pragma unroll
  for (int r = 0; r < 8; ++r) {
    const int grow = rbase + r;
    float v = acc[r] + bb;
    if (mode == 0) {
      v = v > 0.f ? v : 0.2f * v;
      o_erase[grow * MEMD + gcol] = v;
    } else {
      if (gcol < MEMD)          o_erase[grow * MEMD + gcol] = 1.f / (1.f + expf(-v));
      else if (gcol < 2 * MEMD) o_add  [grow * MEMD + gcol - MEMD] = v;
      else                      o_other[grow * MEMD + gcol - 2 * MEMD] = v;
    }
  }
}

// ---------------------------------------------------------------------------
// Fused gate + depthwise 3x3 conv (SAME, cross-correlation) + alpha blend.
// One block per batch row: gate = sigmoid(dot(ghid_row, Wg2) + bg2) via tree
// reduction, then alpha = conv(prev_alpha, k)*gate + prev_alpha*(1-gate).
// ---------------------------------------------------------------------------
__global__ __launch_bounds__(256) void conv_gate_kernel(
    const float* __restrict__ prev_alpha, const float* __restrict__ ghid,
    const float* __restrict__ Wg2, const float* __restrict__ bg2,
    const float* __restrict__ kern9, float* __restrict__ alpha_out)
{
  const int b = blockIdx.x, tid = threadIdx.x;
  __shared__ float red[256];
  __shared__ float s_k[9];
  __shared__ float s_g;

  float pr = ghid[b * MEMD + tid]       * Wg2[tid] +
             ghid[b * MEMD + tid + 256] * Wg2[tid + 256];
  red[tid] = pr; __syncthreads();
  for (int s = 128; s > 0; s >>= 1) { if (tid < s) red[tid] += red[tid + s]; __syncthreads(); }
  if (tid == 0) s_g = 1.f / (1.f + expf(-(red[0] + bg2[0])));
  if (tid < 9) s_k[tid] = kern9[b * 9 + tid];
  __syncthreads();

  const float g = s_g;
  const float* __restrict__ pa = prev_alpha + b * NMEM;
  #pragma unroll
  for (int tpix = 0; tpix < 4; ++tpix) {
    const int pix = tid + tpix * 256;
    const int y = pix >> 5, x = pix & 31;
    float acc = 0.f;
    #pragma unroll
    for (int dy = 0; dy < 3; ++dy) {
      const int yy = y + dy - 1;
      if (yy < 0 || yy > 31) continue;
      #pragma unroll
      for (int dx = 0; dx < 3; ++dx) {
        const int xx = x + dx - 1;
        if (xx < 0 || xx > 31) continue;
        acc += s_k[dy * 3 + dx] * pa[yy * 32 + xx];
      }
    }
    const float pv = pa[pix];
    alpha_out[b * NMEM + pix] = acc * g + pv * (1.f - g);
  }
}

// ---------------------------------------------------------------------------
// Streaming M update + partial read_v (the ~1 GB HBM kernel; bandwidth floor).
// 1024 blocks (b x 4 n-chunks) x 256 threads; float4 granule -> b128
// loads/stores (512 B/wave/instr). Per-block read_v partials go to ws; a tiny
// deterministic reduce kernel finishes (no float atomics -> bit-stable).
// ---------------------------------------------------------------------------
__global__ __launch_bounds__(256) void mupdate_kernel(
    const float* __restrict__ M, const float* __restrict__ alpha,
    const float* __restrict__ erase_v, const float* __restrict__ add_v,
    float* __restrict__ M_out, float* __restrict__ rpart)
{
  const int bx  = blockIdx.x;
  const int b   = bx >> 2;         // batch row
  const int q   = bx & 3;          // n-chunk (256 rows)
  const int tid = threadIdx.x;
  const int nh  = tid >> 7;        // 0..1: which of 2 n-rows per iteration
  const int lid = tid & 127;
  const int d   = lid * 4;         // float4 granule over d
  __shared__ float4 sp[2][128];

  const float4 ev = *(const float4*)(erase_v + b * MEMD + d);
  const float4 av = *(const float4*)(add_v   + b * MEMD + d);
  const float* __restrict__ Mb = M     + (size_t)b * NMEM * MEMD;
  float*       __restrict__ Mo = M_out + (size_t)b * NMEM * MEMD;
  const float* __restrict__ al = alpha + b * NMEM + q * 256;

  float4 r = make_float4(0.f, 0.f, 0.f, 0.f);
  for (int nn = 0; nn < 128; ++nn) {
    const int n = q * 256 + nn * 2 + nh;
    const float a = al[nn * 2 + nh];
    const float4 mv = *(const float4*)(Mb + (size_t)n * MEMD + d);
    if (nn + 2 < 128)
      __builtin_prefetch(Mb + (size_t)(n + 4) * MEMD + d, 0, 1); // global_prefetch_b8
    float4 mn;
    mn.x = mv.x * (1.f - a * ev.x) + a * av.x;
    mn.y = mv.y * (1.f - a * ev.y) + a * av.y;
    mn.z = mv.z * (1.f - a * ev.z) + a * av.z;
    mn.w = mv.w * (1.f - a * ev.w) + a * av.w;
    *(float4*)(Mo + (size_t)n * MEMD + d) = mn;
    r.x += a * mn.x; r.y += a * mn.y; r.z += a * mn.z; r.w += a * mn.w;
  }
  sp[nh][lid] = r;
  __syncthreads();
  if (nh == 0) {
    const float4 t0 = sp[0][lid], t1 = sp[1][lid];
    const float4 t = make_float4(t0.x + t1.x, t0.y + t1.y, t0.z + t1.z, t0.w + t1.w);
    *(float4*)(rpart + (size_t)(b * 4 + q) * MEMD + d) = t;
  }
}

// Deterministic final reduce: read_v[b,d] = sum_q rpart[b,q,d] (fixed order).
__global__ __launch_bounds__(128) void readv_reduce_kernel(
    const float* __restrict__ rpart, float* __restrict__ read_v)
{
  const int b = blockIdx.x;
  const int d = threadIdx.x * 4;
  float4 s = make_float4(0.f, 0.f, 0.f, 0.f);
  #pragma unroll
  for (int q = 0; q < 4; ++q) {
    const float4 t = *(const float4*)(rpart + (size_t)(b * 4 + q) * MEMD + d);
    s.x += t.x; s.y += t.y; s.z += t.z; s.w += t.w;
  }
  *(float4*)(read_v + b * MEMD + d) = s;
}

// ---------------------------------------------------------------------------
extern "C" void kernel_launch(void* const* d_in, const int* in_sizes, int n_in,
                              void* d_out, int out_size, void* d_ws, size_t ws_size,
                              hipStream_t stream) {
  (void)in_sizes; (void)n_in; (void)out_size; (void)ws_size;
  const float* h          = (const float*)d_in[0];
  const float* a          = (const float*)d_in[1];
  /* prev_h (d_in[2]) unused by reference */
  const float* prev_alpha = (const float*)d_in[3];
  const float* M          = (const float*)d_in[4];
  const float* Wv         = (const float*)d_in[5];
  const float* bv         = (const float*)d_in[6];
  const float* Wk1        = (const float*)d_in[7];
  const float* bk1        = (const float*)d_in[8];
  const float* Wk2        = (const float*)d_in[9];
  const float* bk2        = (const float*)d_in[10];
  const float* Wg1        = (const float*)d_in[11];
  const float* bg1        = (const float*)d_in[12];
  const float* Wg2        = (const float*)d_in[13];
  const float* bg2        = (const float*)d_in[14];

  float* out       = (float*)d_out;
  float* read_v    = out + READ_OFF;
  float* other_v   = out + OTHER_OFF;
  float* M_out     = out + M_OFF;
  float* alpha_out = out + ALPHA_OFF;

  float* ws    = (float*)d_ws;          // ~3.7 MB scratch
  float* erase = ws + WS_ERASE;         // 256*512
  float* addv  = ws + WS_ADD;           // 256*512
  float* ghid  = ws + WS_GHID;          // 256*512
  float* kern9 = ws + WS_KERN;          // 256*9
  float* rpart = ws + WS_RPART;         // 256*4*512

  kernelnet_kernel<<<BS, 256, 0, stream>>>(a, Wk1, bk1, Wk2, bk2, kern9);
  // h @ Wg1 (N=512): 512 tiles / 4 waves per block
  gemm_wmma_f32<<<128, 128, 0, stream>>>(h, Wg1, bg1, 512, 0, ghid, nullptr, nullptr);
  // h @ Wv (N=1536): 1536 tiles
  gemm_wmma_f32<<<384, 128, 0, stream>>>(h, Wv, bv, 1536, 1, erase, addv, other_v);
  conv_gate_kernel<<<BS, 256, 0, stream>>>(prev_alpha, ghid, Wg2, bg2, kern9, alpha_out);
  mupdate_kernel<<<BS * 4, 256, 0, stream>>>(M, alpha_out, erase, addv, M_out, rpart);
  readv_reduce_kernel<<<BS, 128, 0, stream>>>(rpart, read_v);
}